// FWHT_84232898609704
// MI455X (gfx1250) — compile-verified
//
#include <hip/hip_runtime.h>

typedef __attribute__((ext_vector_type(2))) float v2f;
typedef __attribute__((ext_vector_type(8))) float v8f;

namespace {
constexpr int ROWLEN = 4096;   // transform length (per row)
constexpr int PAD    = 20;     // padded LDS stride of a 16-float group (bank-conflict free, 16B aligned)
constexpr int REGION = 16 * PAD; // 320 floats per 256-element tile region
}

// Sylvester-order Hadamard entry: H[a][b] = (-1)^popcount(a&b)
__device__ __forceinline__ float hsign(int a, int b) {
    return (__popc(a & b) & 1) ? -1.0f : 1.0f;
}

// One block = one row of 4096 f32. FWHT_4096 = (H16 x H16 x H16) applied as
// three batched 16x16x16 f32 matmuls on the WMMA pipe.
__global__ __launch_bounds__(256) void fwht4096_wmma_kernel(
    const float* __restrict__ in, float* __restrict__ out)
{
    __shared__ float lds1[256 * PAD]; // 5120 floats: padded working buffer
    __shared__ float lds2[ROWLEN];    // 4096 floats: unpadded output staging

    const int tid  = threadIdx.x;
    const int lane = tid & 31;
    const int wave = tid >> 5;   // 8 waves
    const int n    = lane & 15;  // N column within a 16x16 tile (also A row index)
    const int lg   = lane >> 4;  // lane group (0 or 1)

    const float* rowin  = in  + (size_t)blockIdx.x * ROWLEN;
    float*       rowout = out + (size_t)blockIdx.x * ROWLEN;

    // ---- stage in: global -> padded LDS (coalesced b128) ----
#pragma unroll
    for (int q = 0; q < 4; ++q) {
        const int e = q * 1024 + tid * 4;
        const float4 v = *reinterpret_cast<const float4*>(rowin + e);
        const int pe = (e >> 4) * PAD + (e & 15);
        *reinterpret_cast<float4*>(&lds1[pe]) = v;
    }
    __syncthreads();

    // ---- A operand: H16 split into four 16x4 K-chunks ----
    // A layout (16x4 f32): lane m = lane&15; vgpr {x,y} -> K = 4c + 2*lg + {0,1}
    v2f A[4];
#pragma unroll
    for (int c = 0; c < 4; ++c) {
        const int k0 = 4 * c + 2 * lg;
        A[c].x = hsign(n, k0);
        A[c].y = hsign(n, k0 + 1);
    }

    // ================= pass 1: contract fastest axis (k) =================
    // element e = i*256 + j*16 + k ; tile t covers (i,j) with i*16+j = t*16+n
    // B[K][n] = lds1[t*REGION + n*PAD + K]  (transposed read, stride-20 -> conflict free)
#pragma unroll
    for (int tt = 0; tt < 2; ++tt) {
        const int t = 2 * wave + tt;
        const int base = t * REGION;
        v2f B[4];
#pragma unroll
        for (int c = 0; c < 4; ++c) {
            const int k0 = 4 * c + 2 * lg;
            B[c] = *reinterpret_cast<const v2f*>(&lds1[base + n * PAD + k0]);
        }
        v8f acc = {};
#pragma unroll
        for (int c = 0; c < 4; ++c)
            acc = __builtin_amdgcn_wmma_f32_16x16x4_f32(false, A[c], false, B[c],
                                                        (short)0, acc, false, false);
#pragma unroll
        for (int p = 0; p < 8; ++p)
            lds1[base + n * PAD + (p + 8 * lg)] = acc[p];
    }
    __syncthreads();

    // ================= pass 2: contract middle axis (j) =================
    // tile t covers i = t, columns n = k'. B[K=j][n] = lds1[t*REGION + j*PAD + n]
#pragma unroll
    for (int tt = 0; tt < 2; ++tt) {
        const int t = 2 * wave + tt;
        const int base = t * REGION;
        v2f B[4];
#pragma unroll
        for (int c = 0; c < 4; ++c) {
            const int k0 = 4 * c + 2 * lg;
            B[c].x = lds1[base + k0 * PAD + n];
            B[c].y = lds1[base + (k0 + 1) * PAD + n];
        }
        v8f acc = {};
#pragma unroll
        for (int c = 0; c < 4; ++c)
            acc = __builtin_amdgcn_wmma_f32_16x16x4_f32(false, A[c], false, B[c],
                                                        (short)0, acc, false, false);
#pragma unroll
        for (int p = 0; p < 8; ++p)
            lds1[base + (p + 8 * lg) * PAD + n] = acc[p];
    }
    __syncthreads();

    // ================= pass 3: contract outer axis (i) =================
    // tile t covers j' = t, columns n = k'. B[K=i][n] = lds1[i*REGION + t*PAD + n]
    // D goes to unpadded lds2 at i'*256 + t*16 + n (== final element index)
#pragma unroll
    for (int tt = 0; tt < 2; ++tt) {
        const int t = 2 * wave + tt;
        v2f B[4];
#pragma unroll
        for (int c = 0; c < 4; ++c) {
            const int k0 = 4 * c + 2 * lg;
            B[c].x = lds1[k0 * REGION + t * PAD + n];
            B[c].y = lds1[(k0 + 1) * REGION + t * PAD + n];
        }
        v8f acc = {};
#pragma unroll
        for (int c = 0; c < 4; ++c)
            acc = __builtin_amdgcn_wmma_f32_16x16x4_f32(false, A[c], false, B[c],
                                                        (short)0, acc, false, false);
#pragma unroll
        for (int p = 0; p < 8; ++p)
            lds2[(p + 8 * lg) * 256 + t * 16 + n] = acc[p];
    }
    __syncthreads();

    // ---- stage out: LDS -> global (coalesced b128) ----
#pragma unroll
    for (int q = 0; q < 4; ++q) {
        const int e = q * 1024 + tid * 4;
        *reinterpret_cast<float4*>(rowout + e) =
            *reinterpret_cast<const float4*>(&lds2[e]);
    }
}

extern "C" void kernel_launch(void* const* d_in, const int* in_sizes, int n_in,
                              void* d_out, int out_size, void* d_ws, size_t ws_size,
                              hipStream_t stream) {
    const float* x = (const float*)d_in[0];
    float* y = (float*)d_out;
    const int rows = in_sizes[0] / ROWLEN;   // 8192 for the reference shapes
    fwht4096_wmma_kernel<<<dim3(rows), dim3(256), 0, stream>>>(x, y);
}